// CoVisFormer_15882789061209
// MI455X (gfx1250) — compile-verified
//
#include <hip/hip_runtime.h>
#include <hip/hip_bf16.h>

typedef __attribute__((ext_vector_type(2))) float v2f;
typedef __attribute__((ext_vector_type(8))) float v8f;

static __device__ __forceinline__ v8f wmma_f32(v2f a, v2f b, v8f c) {
  // D(16x16,f32) = A(16x4,f32) * B(4x16,f32) + C
  return __builtin_amdgcn_wmma_f32_16x16x4_f32(
      /*neg_a=*/false, a, /*neg_b=*/false, b,
      /*c_mod=*/(short)0, c, /*reuse_a=*/false, /*reuse_b=*/false);
}

static __device__ __forceinline__ float wave_reduce_add(float v) {
#pragma unroll
  for (int m = 1; m < 32; m <<= 1) v += __shfl_xor(v, m, 32);
  return v;
}

// ---------------------------------------------------------------------------
// Kernel 1: LayerNorm fused with rearrange 'b l x y w1 w2 d -> (b x y)(l w1 w2) d'
// One wave per 256-wide row.  81920 rows total.
// ---------------------------------------------------------------------------
__global__ __launch_bounds__(256) void ln_x_kernel(
    const float* __restrict__ x, const float* __restrict__ gw,
    const float* __restrict__ gb, float* __restrict__ xn) {
  const int row  = blockIdx.x * 8 + (threadIdx.x >> 5);
  const int lane = threadIdx.x & 31;
  const int Bi = row / 320, n = row % 320;
  const int b  = Bi >> 6, hx = (Bi >> 3) & 7, wy = Bi & 7;
  const int l  = n >> 6,  s1 = (n >> 3) & 7,  s2 = n & 7;
  const size_t src =
      ((((((size_t)b * 5 + l) * 8 + hx) * 8 + wy) * 8 + s1) * 8 + s2) * 256;

  const float4* p = (const float4*)(x + src);
  float4 a = p[lane];
  float4 c = p[32 + lane];
  float s = a.x + a.y + a.z + a.w + c.x + c.y + c.z + c.w;
  float q = a.x*a.x + a.y*a.y + a.z*a.z + a.w*a.w
          + c.x*c.x + c.y*c.y + c.z*c.z + c.w*c.w;
  s = wave_reduce_add(s);
  q = wave_reduce_add(q);
  const float mean = s * (1.0f / 256.0f);
  const float var  = q * (1.0f / 256.0f) - mean * mean;
  const float rstd = rsqrtf(var + 1e-5f);

  const float4* wp = (const float4*)gw;
  const float4* bp = (const float4*)gb;
  float4 w0 = wp[lane], w1 = wp[32 + lane];
  float4 b0 = bp[lane], b1 = bp[32 + lane];
  float4 o0, o1;
  o0.x = (a.x - mean) * rstd * w0.x + b0.x;
  o0.y = (a.y - mean) * rstd * w0.y + b0.y;
  o0.z = (a.z - mean) * rstd * w0.z + b0.z;
  o0.w = (a.w - mean) * rstd * w0.w + b0.w;
  o1.x = (c.x - mean) * rstd * w1.x + b1.x;
  o1.y = (c.y - mean) * rstd * w1.y + b1.y;
  o1.z = (c.z - mean) * rstd * w1.z + b1.z;
  o1.w = (c.w - mean) * rstd * w1.w + b1.w;
  float4* o = (float4*)(xn + (size_t)row * 256);
  o[lane] = o0;
  o[32 + lane] = o1;
}

// ---------------------------------------------------------------------------
// Kernel: plain row LayerNorm, in place.  One wave per 256-wide row.
// ---------------------------------------------------------------------------
__global__ __launch_bounds__(256) void ln_rows_kernel(
    float* __restrict__ data, const float* __restrict__ gw,
    const float* __restrict__ gb) {
  const int row  = blockIdx.x * 8 + (threadIdx.x >> 5);
  const int lane = threadIdx.x & 31;
  float4* p = (float4*)(data + (size_t)row * 256);
  float4 a = p[lane];
  float4 c = p[32 + lane];
  float s = a.x + a.y + a.z + a.w + c.x + c.y + c.z + c.w;
  float q = a.x*a.x + a.y*a.y + a.z*a.z + a.w*a.w
          + c.x*c.x + c.y*c.y + c.z*c.z + c.w*c.w;
  s = wave_reduce_add(s);
  q = wave_reduce_add(q);
  const float mean = s * (1.0f / 256.0f);
  const float var  = q * (1.0f / 256.0f) - mean * mean;
  const float rstd = rsqrtf(var + 1e-5f);
  const float4* wp = (const float4*)gw;
  const float4* bp = (const float4*)gb;
  float4 w0 = wp[lane], w1 = wp[32 + lane];
  float4 b0 = bp[lane], b1 = bp[32 + lane];
  float4 o0, o1;
  o0.x = (a.x - mean) * rstd * w0.x + b0.x;
  o0.y = (a.y - mean) * rstd * w0.y + b0.y;
  o0.z = (a.z - mean) * rstd * w0.z + b0.z;
  o0.w = (a.w - mean) * rstd * w0.w + b0.w;
  o1.x = (c.x - mean) * rstd * w1.x + b1.x;
  o1.y = (c.y - mean) * rstd * w1.y + b1.y;
  o1.z = (c.z - mean) * rstd * w1.z + b1.z;
  o1.w = (c.w - mean) * rstd * w1.w + b1.w;
  p[lane] = o0;
  p[32 + lane] = o1;
}

// ---------------------------------------------------------------------------
// Generic fp32 WMMA GEMM: C(MxN) = A(MxK) @ B(KxN).  Each wave: 16x32 tile.
// ---------------------------------------------------------------------------
__global__ __launch_bounds__(128) void gemm_kernel(
    const float* __restrict__ A, int lda, const float* __restrict__ Bm, int ldb,
    float* __restrict__ C, int ldc, int mtiles, int npairs, int K) {
  const int wid  = blockIdx.x * 4 + (threadIdx.x >> 5);
  const int mt   = wid / npairs, np = wid % npairs;
  if (mt >= mtiles) return;
  const int lane = threadIdx.x & 31;
  const int half = lane >> 4, c = lane & 15;
  const int m0 = mt * 16, n0 = np * 32;

  const float* arow = A + (size_t)(m0 + c) * lda + 2 * half;
  const float* bcol = Bm + (size_t)(2 * half) * ldb + n0 + c;
  v8f acc0 = {}, acc1 = {};
  for (int kk = 0; kk < K; kk += 4) {
    v2f a = *(const v2f*)(arow + kk);
    const float* bp = bcol + (size_t)kk * ldb;
    v2f b0, b1;
    b0.x = bp[0];  b0.y = bp[ldb];
    b1.x = bp[16]; b1.y = bp[ldb + 16];
    acc0 = wmma_f32(a, b0, acc0);
    acc1 = wmma_f32(a, b1, acc1);
  }
  float* crow = C + (size_t)(m0 + 8 * half) * ldc + n0 + c;
#pragma unroll
  for (int j = 0; j < 8; ++j) {
    crow[(size_t)j * ldc]      = acc0[j];
    crow[(size_t)j * ldc + 16] = acc1[j];
  }
}

// ---------------------------------------------------------------------------
// Spatial-reduction conv as 4 accumulating K=256 GEMM passes (one per kh,kw).
// out[b, m, o] = sum_{ci,kh,kw} xn[b, n(m,kh,kw), ci] * sr_w[o, ci, kh, kw] + sr_b[o]
// m in [0,80): i'=m/20, j'=m%20, n=(2i'+kh)*40 + 2j'+kw.  Wave: 16x32 tile.
// ---------------------------------------------------------------------------
__global__ __launch_bounds__(128) void sr_conv_kernel(
    const float* __restrict__ xn, const float* __restrict__ srw,
    const float* __restrict__ srb, float* __restrict__ out) {
  const int wid = blockIdx.x * 4 + (threadIdx.x >> 5);  // 0..10239
  const int b = wid / 40;
  const int rem = wid % 40;
  const int mt = rem / 8, np = rem % 8;
  const int lane = threadIdx.x & 31;
  const int half = lane >> 4, c = lane & 15;
  const int m0 = mt * 16, n0 = np * 32;
  const int rowm = m0 + c;                 // 0..79
  const int i2 = rowm / 20, j2 = rowm % 20;

  v8f acc0 = {}, acc1 = {};
#pragma unroll
  for (int kh = 0; kh < 2; ++kh) {
#pragma unroll
    for (int kw = 0; kw < 2; ++kw) {
      const int nIdx = (2 * i2 + kh) * 40 + 2 * j2 + kw;
      const float* arow = xn + ((size_t)b * 320 + nIdx) * 256 + 2 * half;
      // B[kci][o] = srw[o*1024 + kci*4 + kh*2 + kw]
      const float* w0 = srw + (size_t)(n0 + c) * 1024 + (2 * half) * 4 + kh * 2 + kw;
      const float* w1 = w0 + (size_t)16 * 1024;
      for (int kk = 0; kk < 256; kk += 4) {
        v2f a = *(const v2f*)(arow + kk);
        const float* wp = w0 + (size_t)kk * 4;
        const float* wq = w1 + (size_t)kk * 4;
        v2f b0, b1;
        b0.x = wp[0]; b0.y = wp[4];
        b1.x = wq[0]; b1.y = wq[4];
        acc0 = wmma_f32(a, b0, acc0);
        acc1 = wmma_f32(a, b1, acc1);
      }
    }
  }
  const float bias0 = srb[n0 + c], bias1 = srb[n0 + 16 + c];
  float* crow = out + ((size_t)b * 80 + m0 + 8 * half) * 256 + n0 + c;
#pragma unroll
  for (int j = 0; j < 8; ++j) {
    crow[(size_t)j * 256]      = acc0[j] + bias0;
    crow[(size_t)j * 256 + 16] = acc1[j] + bias1;
  }
}

// ---------------------------------------------------------------------------
// Fused attention per (b,h,mtile): scores = q@k^T * 0.125, softmax over 80,
// out = attn @ v.  Each wave handles 16 query rows; scores staged in LDS.
// q layout: (B,320,512) col h*64+d ; kv layout: (B,80,1024), k=[0,512) v=[512,1024)
// ---------------------------------------------------------------------------
__global__ __launch_bounds__(128) void attn_kernel(
    const float* __restrict__ q, const float* __restrict__ kv,
    float* __restrict__ out) {
  __shared__ float smem[4 * 16 * 80];
  const int wid  = blockIdx.x * 4 + (threadIdx.x >> 5);  // 0..40959
  const int lane = threadIdx.x & 31;
  const int half = lane >> 4, c = lane & 15;
  const int mt = wid % 20;
  const int bh = wid / 20;
  const int h = bh & 7, b = bh >> 3;
  const int m0 = mt * 16;
  float* ws = smem + (threadIdx.x >> 5) * (16 * 80);

  // ---- scores (16 x 80), scaled ----
  const float* qrow = q + ((size_t)b * 320 + m0 + c) * 512 + h * 64 + 2 * half;
#pragma unroll
  for (int nt = 0; nt < 5; ++nt) {
    v8f acc = {};
    const float* krow = kv + ((size_t)b * 80 + nt * 16 + c) * 1024 + h * 64 + 2 * half;
    for (int kk = 0; kk < 64; kk += 4) {
      v2f a  = *(const v2f*)(qrow + kk);
      v2f bb = *(const v2f*)(krow + kk);   // k^T frag: contiguous along feature dim
      acc = wmma_f32(a, bb, acc);
    }
#pragma unroll
    for (int j = 0; j < 8; ++j)
      ws[(j + 8 * half) * 80 + nt * 16 + c] = acc[j] * 0.125f;
  }
  __syncthreads();

  // ---- row softmax over 80: lane (half,c) owns row c, columns [half*40, half*40+40)
  {
    float* rowp = ws + c * 80 + half * 40;
    float mx = -3.4e38f;
    for (int i = 0; i < 40; ++i) mx = fmaxf(mx, rowp[i]);
    mx = fmaxf(mx, __shfl_xor(mx, 16, 32));
    float s = 0.0f;
    for (int i = 0; i < 40; ++i) { float e = __expf(rowp[i] - mx); rowp[i] = e; s += e; }
    s += __shfl_xor(s, 16, 32);
    const float inv = 1.0f / s;
    for (int i = 0; i < 40; ++i) rowp[i] *= inv;
  }
  __syncthreads();

  // ---- out = attn(16x80) @ v(80x64) ----
  const float* afrag = ws + c * 80 + 2 * half;   // A-frag straight from LDS
#pragma unroll
  for (int nt = 0; nt < 4; ++nt) {
    v8f acc = {};
    const float* vcol = kv + ((size_t)b * 80 + 2 * half) * 1024 + 512 + h * 64 + nt * 16 + c;
    for (int kk = 0; kk < 80; kk += 4) {
      v2f a = *(const v2f*)(afrag + kk);
      const float* vp = vcol + (size_t)kk * 1024;
      v2f bb;
      bb.x = vp[0]; bb.y = vp[1024];
      acc = wmma_f32(a, bb, acc);
    }
    float* crow = out + ((size_t)b * 320 + m0 + 8 * half) * 512 + h * 64 + nt * 16 + c;
#pragma unroll
    for (int j = 0; j < 8; ++j) crow[(size_t)j * 512] = acc[j];
  }
}

// ---------------------------------------------------------------------------
// Final projection (81920x512)@(512x64) fused with scatter to
// (b, l, x, y, w1, w2, 64).  Each wave: 16x32 tile; npairs = 2.
// ---------------------------------------------------------------------------
__global__ __launch_bounds__(128) void outproj_kernel(
    const float* __restrict__ A, const float* __restrict__ Wo,
    float* __restrict__ out) {
  const int wid = blockIdx.x * 4 + (threadIdx.x >> 5);  // 0..10239
  const int lane = threadIdx.x & 31;
  const int half = lane >> 4, c = lane & 15;
  const int np = wid & 1, mt = wid >> 1;
  const int m0 = mt * 16, n0 = np * 32;

  const float* arow = A + (size_t)(m0 + c) * 512 + 2 * half;
  const float* bcol = Wo + (size_t)(2 * half) * 64 + n0 + c;
  v8f acc0 = {}, acc1 = {};
  for (int kk = 0; kk < 512; kk += 4) {
    v2f a = *(const v2f*)(arow + kk);
    const float* bp = bcol + (size_t)kk * 64;
    v2f b0, b1;
    b0.x = bp[0];  b0.y = bp[64];
    b1.x = bp[16]; b1.y = bp[80];
    acc0 = wmma_f32(a, b0, acc0);
    acc1 = wmma_f32(a, b1, acc1);
  }
#pragma unroll
  for (int j = 0; j < 8; ++j) {
    const int r = m0 + j + 8 * half;
    const int Bi = r / 320, n = r % 320;
    const int b = Bi >> 6, hx = (Bi >> 3) & 7, wy = Bi & 7;
    const int l = n >> 6,  s1 = (n >> 3) & 7, s2 = n & 7;
    const size_t base =
        ((((((size_t)b * 5 + l) * 8 + hx) * 8 + wy) * 8 + s1) * 8 + s2) * 64;
    out[base + n0 + c]      = acc0[j];
    out[base + n0 + 16 + c] = acc1[j];
  }
}

// ---------------------------------------------------------------------------
extern "C" void kernel_launch(void* const* d_in, const int* in_sizes, int n_in,
                              void* d_out, int out_size, void* d_ws, size_t ws_size,
                              hipStream_t stream) {
  const float* x      = (const float*)d_in[0];
  const float* norm_w = (const float*)d_in[1];
  const float* norm_b = (const float*)d_in[2];
  const float* wq     = (const float*)d_in[3];
  const float* sr_w   = (const float*)d_in[4];
  const float* sr_b   = (const float*)d_in[5];
  const float* srn_w  = (const float*)d_in[6];
  const float* srn_b  = (const float*)d_in[7];
  const float* wkv    = (const float*)d_in[8];
  const float* wout   = (const float*)d_in[9];
  float* out = (float*)d_out;

  float* xn       = (float*)d_ws;                    // 81920 x 256
  float* qbuf     = xn     + (size_t)81920 * 256;    // 81920 x 512
  float* srbuf    = qbuf   + (size_t)81920 * 512;    // 20480 x 256
  float* kvbuf    = srbuf  + (size_t)20480 * 256;    // 20480 x 1024
  float* attn_out = kvbuf  + (size_t)20480 * 1024;   // 81920 x 512

  // 1) LN + rearrange:  x -> xn (81920 x 256)
  ln_x_kernel<<<10240, 256, 0, stream>>>(x, norm_w, norm_b, xn);
  // 2) q = xn @ wq : (81920x256)(256x512)
  gemm_kernel<<<20480, 128, 0, stream>>>(xn, 256, wq, 512, qbuf, 512,
                                         /*mtiles=*/5120, /*npairs=*/16, /*K=*/256);
  // 3) spatial-reduction conv -> srbuf (B,80,256) with bias
  sr_conv_kernel<<<2560, 128, 0, stream>>>(xn, sr_w, sr_b, srbuf);
  // 4) LN on srbuf (in place), 20480 rows
  ln_rows_kernel<<<2560, 256, 0, stream>>>(srbuf, srn_w, srn_b);
  // 5) kv = srbuf @ wkv : (20480x256)(256x1024)
  gemm_kernel<<<10240, 128, 0, stream>>>(srbuf, 256, wkv, 1024, kvbuf, 1024,
                                         /*mtiles=*/1280, /*npairs=*/32, /*K=*/256);
  // 6) fused attention -> attn_out (81920 x 512)
  attn_kernel<<<10240, 128, 0, stream>>>(qbuf, kvbuf, attn_out);
  // 7) out-projection + scatter to final layout
  outproj_kernel<<<2560, 128, 0, stream>>>(attn_out, wout, out);
}